// ColorMatchGrads_3728031613261
// MI455X (gfx1250) — compile-verified
//
#include <hip/hip_runtime.h>
#include <hip/hip_bf16.h>
#include <math.h>

#define NPIX     (512*512)
#define NBINS    255
#define NBINSP   256      // padded bins; bin 255 always 0
#define HISTROWS 16       // padded image rows for branch-free WMMA A/B loads
#define NWAVES   8        // 256-thread blocks = 8 wave32

typedef float v2f __attribute__((ext_vector_type(2)));
typedef float v8f __attribute__((ext_vector_type(8)));

// ---------------------------------------------------------------------------
// Shared forward math (matches JAX reference semantics, incl. clip selects)
// ---------------------------------------------------------------------------
struct Fwd {
    float r, g, b;        // clipped rgb
    float yr, yg, yb;     // pre-clip (img+1)*0.5
    float maxc, minc, dc; // max/min/delta
    float dd;             // where(dc==0, 1, dc)
    float sc, vc;         // clipped s, v
    float w;              // sqrt(s*v)
    float x;              // clipped hue
    float binlo, binhi;
    int   idx, branch;
    bool  valid;
};

__device__ __forceinline__ Fwd fwd_pixel(float ir, float ig, float ib) {
    Fwd f;
    const float E  = 1e-8f;
    const float HI = 1.0f - 1e-8f;   // folds to 1.0f in f32, same as JAX
    f.yr = (ir + 1.f) * 0.5f;
    f.yg = (ig + 1.f) * 0.5f;
    f.yb = (ib + 1.f) * 0.5f;
    f.r = fminf(fmaxf(f.yr, E), HI);
    f.g = fminf(fmaxf(f.yg, E), HI);
    f.b = fminf(fmaxf(f.yb, E), HI);
    f.maxc = fmaxf(f.r, fmaxf(f.g, f.b));
    f.minc = fminf(f.r, fminf(f.g, f.b));
    f.dc   = f.maxc - f.minc;
    float s = f.dc / fmaxf(f.maxc, E);
    f.dd = (f.dc == 0.f) ? 1.f : f.dc;
    float hraw;
    if (f.maxc == f.r)      { f.branch = 0; hraw = (f.g - f.b) / f.dd; }
    else if (f.maxc == f.g) { f.branch = 1; hraw = 2.f + (f.b - f.r) / f.dd; }
    else                    { f.branch = 2; hraw = 4.f + (f.r - f.g) / f.dd; }
    float h6 = hraw * (1.f / 6.f);
    float hm = h6 - floorf(h6);        // x % 1.0
    float h  = (f.dc == 0.f) ? 0.f : hm;
    f.x  = fminf(fmaxf(h, 0.f), 1.f);
    f.sc = fminf(fmaxf(s, 0.f), 1.f);
    f.vc = fminf(fmaxf(f.maxc, 0.f), 1.f);
    f.w  = sqrtf(f.sc * f.vc);
    const float delta = 1.f / 254.f;
    int idx = (int)floorf(f.x / delta);
    idx = idx < 0 ? 0 : (idx > NBINS - 1 ? NBINS - 1 : idx);
    f.idx   = idx;
    f.valid = (idx + 1) < NBINS;
    f.binlo = idx * delta;
    f.binhi = (idx + 1) * delta;
    return f;
}

// ---------------------------------------------------------------------------
// Kernel 0: zero padded per-image histograms (16 x 256 floats)
// ---------------------------------------------------------------------------
__global__ void k_zero(float* __restrict__ p, int n) {
    int i = blockIdx.x * blockDim.x + threadIdx.x;
    if (i < n) p[i] = 0.f;
}

// ---------------------------------------------------------------------------
// Kernel 1: per-image soft histogram, float4 (b128) loads, per-wave private
// LDS histograms (ds_add_f32), branch-free merge + global atomics.
// grid = (chunks, B), block = 256
// ---------------------------------------------------------------------------
__device__ __forceinline__ void hist_accum(float* lh, float ir, float ig, float ib) {
    Fwd f = fwd_pixel(ir, ig, ib);
    float lo = (f.binhi - f.x) * f.w;
    float hi = (f.x - f.binlo) * f.w;
    atomicAdd(&lh[f.idx], lo);
    if (f.valid) atomicAdd(&lh[f.idx + 1], hi);
}

__global__ void k_hist(const float* __restrict__ img, float* __restrict__ hist,
                       int npix) {
    __shared__ float lh[NWAVES][NBINSP];
    const int b    = blockIdx.y;
    const int tid  = threadIdx.x;
    const int wave = tid >> 5;
    for (int t = tid; t < NWAVES * NBINSP; t += blockDim.x)
        ((float*)lh)[t] = 0.f;
    __syncthreads();

    const float*  base = img + (size_t)b * 3 * npix;
    const float4* r4 = (const float4*)(base);
    const float4* g4 = (const float4*)(base + npix);
    const float4* b4 = (const float4*)(base + 2 * npix);
    const int nvec = npix >> 2;
    float* lw = lh[wave];
    for (int i = blockIdx.x * blockDim.x + tid; i < nvec;
         i += gridDim.x * blockDim.x) {
        float4 vr = r4[i], vg = g4[i], vb = b4[i];
        hist_accum(lw, vr.x, vg.x, vb.x);
        hist_accum(lw, vr.y, vg.y, vb.y);
        hist_accum(lw, vr.z, vg.z, vb.z);
        hist_accum(lw, vr.w, vg.w, vb.w);
    }
    __syncthreads();
    // branch-free merge of the 8 wave-private copies for this thread's bin
    float sum = 0.f;
#pragma unroll
    for (int wv = 0; wv < NWAVES; ++wv) sum += lh[wv][tid];
    atomicAdd(&hist[b * NBINSP + tid], sum);
}

// ---------------------------------------------------------------------------
// Kernel 2: one wave (32 lanes). Chained V_WMMA_F32_16X16X4_F32, all operand
// loads branch-free (hist padded to 16 zero rows; target staged zero-padded
// in LDS):
//   chain 1: A rows = hist rows, B = ones   -> D[i][*]  = S_i   (exact f32)
//   chain 2: A rows = g rows, B cols = hist -> diag(D)  = sum_k g*H
// Then G[b][k] = (g_k - dot_b/S_b)/S_b  (grad w.r.t. raw hist bins).
// ---------------------------------------------------------------------------
__global__ void k_bins(const float* __restrict__ hist,
                       const float* __restrict__ target,
                       float* __restrict__ G, int B, float cmean) {
    __shared__ float sS[HISTROWS];
    __shared__ float sD[8];
    __shared__ float sT[HISTROWS * NBINSP];   // zero-padded target
    const int lane = threadIdx.x;
    const int row  = lane & 15;              // A-matrix M / B-matrix N index
    const int koff = (lane < 16) ? 0 : 2;    // 32-bit A 16x4: lanes16-31 hold K=2,3
    const int rbase = row * NBINSP;

    // stage target, zero-padded to 16 x 256
    for (int t = lane; t < HISTROWS * NBINSP; t += 32) {
        int bb = t >> 8, kk = t & 255;
        sT[t] = (bb < B && kk < NBINS) ? target[bb * NBINS + kk] : 0.f;
    }
    __syncthreads();

    const float2* hist2 = (const float2*)hist;   // k0 is always even

    // ---- WMMA chain 1: per-image histogram sums -------------------------
    v8f c1 = {};
    v2f ones; ones.x = 1.f; ones.y = 1.f;
    for (int kc = 0; kc < 64; ++kc) {
        int k0 = 4 * kc + koff;
        float2 h01 = hist2[(rbase + k0) >> 1];
        v2f a; a.x = h01.x; a.y = h01.y;
        c1 = __builtin_amdgcn_wmma_f32_16x16x4_f32(
                 false, a, false, ones, (short)0, c1, false, false);
    }
    // lane 0 holds D[r][0] = S_r in c1[r]; rows >= B get S = 1 (their H == 0)
    if (lane == 0) {
        sS[0] = (0 < B) ? c1[0] : 1.f;  sS[1] = (1 < B) ? c1[1] : 1.f;
        sS[2] = (2 < B) ? c1[2] : 1.f;  sS[3] = (3 < B) ? c1[3] : 1.f;
        sS[4] = (4 < B) ? c1[4] : 1.f;  sS[5] = (5 < B) ? c1[5] : 1.f;
        sS[6] = (6 < B) ? c1[6] : 1.f;  sS[7] = (7 < B) ? c1[7] : 1.f;
        sS[8]  = 1.f; sS[9]  = 1.f; sS[10] = 1.f; sS[11] = 1.f;
        sS[12] = 1.f; sS[13] = 1.f; sS[14] = 1.f; sS[15] = 1.f;
    }
    __syncthreads();

    // ---- WMMA chain 2: dot_b = sum_k g[b,k] * H[b,k] (diagonal of D) ----
    float S0 = sS[row];
    v8f c2 = {};
    for (int kc = 0; kc < 64; ++kc) {
        int k0 = 4 * kc + koff;
        float2 h01 = hist2[(rbase + k0) >> 1];
        float t0 = sT[rbase + k0], t1 = sT[rbase + k0 + 1];
        v2f a, bb;
        a.x = cmean * (h01.x / S0 - t0);      // g rows (0 for padded rows)
        a.y = cmean * (h01.y / S0 - t1);
        bb.x = h01.x;                         // B[k][col], col == row
        bb.y = h01.y;
        c2 = __builtin_amdgcn_wmma_f32_16x16x4_f32(
                 false, a, false, bb, (short)0, c2, false, false);
    }
    // diag: lane b holds D[b][b] in c2[b] (b < 8)
    if (lane == 0) sD[0] = c2[0];
    if (lane == 1) sD[1] = c2[1];
    if (lane == 2) sD[2] = c2[2];
    if (lane == 3) sD[3] = c2[3];
    if (lane == 4) sD[4] = c2[4];
    if (lane == 5) sD[5] = c2[5];
    if (lane == 6) sD[6] = c2[6];
    if (lane == 7) sD[7] = c2[7];
    __syncthreads();

    // ---- grad w.r.t. raw histogram bins ---------------------------------
    for (int k = lane; k < NBINSP; k += 32) {
        for (int b = 0; b < B; ++b) {
            float S  = sS[b];
            float Hk = hist[b * NBINSP + k];
            float tg = sT[b * NBINSP + k];
            float gk = cmean * (Hk / S - tg);
            float Gv = (k < NBINS) ? (gk - sD[b] / S) / S : 0.f;
            G[b * NBINSP + k] = Gv;
        }
    }
}

// ---------------------------------------------------------------------------
// Per-pixel backward (chain through hue/sat/val and clips)
// ---------------------------------------------------------------------------
__device__ __forceinline__ void grad_pixel(const float* sG, float ir, float ig,
                                           float ib, float& dr, float& dg,
                                           float& db) {
    Fwd f = fwd_pixel(ir, ig, ib);
    float G0 = sG[f.idx];
    float G1 = f.valid ? sG[f.idx + 1] : 0.f;

    float dLdx = f.w * (G1 - G0);
    float dLdw = G0 * (f.binhi - f.x) + G1 * (f.x - f.binlo);

    float dLdhm   = (f.dc == 0.f) ? 0.f : dLdx;     // where(dc==0, 0, .)
    float dLdhraw = dLdhm * (1.f / 6.f);

    float dLdu = dLdw * 0.5f / f.w;     // inf at u==0 absorbed by clip selects
    float dLds = dLdu * f.vc;
    float dLdv = dLdu * f.sc;

    float m        = fmaxf(f.maxc, 1e-8f);
    float dLddc_s  = dLds / m;
    float dLdmax_s = -dLds * f.dc / (m * m);

    float inv = dLdhraw / f.dd;
    float dd2 = f.dd * f.dd;
    float hr = 0.f, hg = 0.f, hb = 0.f, dLddc_h = 0.f;
    if (f.branch == 0)      { hg =  inv; hb = -inv; dLddc_h = -dLdhraw * (f.g - f.b) / dd2; }
    else if (f.branch == 1) { hb =  inv; hr = -inv; dLddc_h = -dLdhraw * (f.b - f.r) / dd2; }
    else                    { hr =  inv; hg = -inv; dLddc_h = -dLdhraw * (f.r - f.g) / dd2; }

    float dLddc = dLddc_s + ((f.dc == 0.f) ? 0.f : dLddc_h);
    float Gmax = dLdv + dLdmax_s + dLddc;
    float Gmin = -dLddc;

    float pr = hr, pg = hg, pb = hb;
    if (f.maxc == f.r)      pr += Gmax;
    else if (f.maxc == f.g) pg += Gmax;
    else                    pb += Gmax;
    if (f.minc == f.r)      pr += Gmin;
    else if (f.minc == f.g) pg += Gmin;
    else                    pb += Gmin;

    const float E = 1e-8f, HI = 1.0f - 1e-8f;
    dr = (f.yr > E && f.yr < HI) ? pr * 0.5f : 0.f;
    dg = (f.yg > E && f.yg < HI) ? pg * 0.5f : 0.f;
    db = (f.yb > E && f.yb < HI) ? pb * 0.5f : 0.f;
}

// ---------------------------------------------------------------------------
// Kernel 3: float4 loads/stores, G staged in LDS.
// grid = (npix/(4*256), B), block = 256 — exact, no tail.
// ---------------------------------------------------------------------------
__global__ void k_grad(const float* __restrict__ img,
                       const float* __restrict__ G,
                       float* __restrict__ out, int npix) {
    __shared__ float sG[NBINSP];
    const int b   = blockIdx.y;
    const int tid = threadIdx.x;
    sG[tid] = G[b * NBINSP + tid];
    __syncthreads();

    const int nvec = npix >> 2;
    int i = blockIdx.x * blockDim.x + tid;
    if (i >= nvec) return;

    const float* base = img + (size_t)b * 3 * npix;
    float4 vr = ((const float4*)(base))[i];
    float4 vg = ((const float4*)(base + npix))[i];
    float4 vb = ((const float4*)(base + 2 * npix))[i];

    float4 orr, og, ob;
    grad_pixel(sG, vr.x, vg.x, vb.x, orr.x, og.x, ob.x);
    grad_pixel(sG, vr.y, vg.y, vb.y, orr.y, og.y, ob.y);
    grad_pixel(sG, vr.z, vg.z, vb.z, orr.z, og.z, ob.z);
    grad_pixel(sG, vr.w, vg.w, vb.w, orr.w, og.w, ob.w);

    float* obase = out + (size_t)b * 3 * npix;
    ((float4*)(obase))[i]            = orr;
    ((float4*)(obase + npix))[i]     = og;
    ((float4*)(obase + 2 * npix))[i] = ob;
}

// ---------------------------------------------------------------------------
extern "C" void kernel_launch(void* const* d_in, const int* in_sizes, int n_in,
                              void* d_out, int out_size, void* d_ws, size_t ws_size,
                              hipStream_t stream) {
    const float* img    = (const float*)d_in[0];
    const float* target = (const float*)d_in[1];
    // d_in[2] is `t` — unused by the reference gradient.

    const int npix = NPIX;
    int B = in_sizes[0] / (3 * npix);
    if (B > 8) B = 8;   // reference uses B == 8 (WMMA packs 8 rows)

    float* hist = (float*)d_ws;                    // 16 * 256 (zero-padded)
    float* G    = hist + HISTROWS * NBINSP;        // B * 256
    float cmean = 2.0f * 1000.0f / (float)(B * NBINS);   // 2*SCALE/(B*NBINS)

    int nz = HISTROWS * NBINSP;
    k_zero<<<(nz + 255) / 256, 256, 0, stream>>>(hist, nz);

    dim3 g1(64, B);
    k_hist<<<g1, 256, 0, stream>>>(img, hist, npix);

    k_bins<<<1, 32, 0, stream>>>(hist, target, G, B, cmean);

    dim3 g3(((npix >> 2) + 255) / 256, B);
    k_grad<<<g3, 256, 0, stream>>>(img, G, (float*)d_out, npix);
}